// MaskedGraphAttentionBlock_45810121179263
// MI455X (gfx1250) — compile-verified
//
#include <hip/hip_runtime.h>
#include <math.h>

typedef float v2f __attribute__((ext_vector_type(2)));
typedef float v8f __attribute__((ext_vector_type(8)));

#define CC 256
#define LDA 260      // padded LDS stride for 256-wide A tiles (260%64=4 -> rows hit distinct banks)
#define LDG 1028     // padded LDS stride for 1024-wide gelu tile
#define EPSV 1e-5f

__device__ __forceinline__ v8f wmma_f32(v2f a, v2f b, v8f c) {
    return __builtin_amdgcn_wmma_f32_16x16x4_f32(false, a, false, b, (short)0, c, false, false);
}

// ---------------- Kernel 1: LayerNorm + QKV projections ----------------
// grid = M/16 blocks, 256 threads (8 waves). Each block: 16 tokens.
__global__ void ln_qkv_kernel(const float* __restrict__ x,
                              const float* __restrict__ nw, const float* __restrict__ nb,
                              const float* __restrict__ wq, const float* __restrict__ bq,
                              const float* __restrict__ wk, const float* __restrict__ bk,
                              const float* __restrict__ wv, const float* __restrict__ bv,
                              float* __restrict__ q, float* __restrict__ k, float* __restrict__ v) {
    extern __shared__ float sA[];                 // [16][LDA]
    const int tid = threadIdx.x;
    const int wave = tid >> 5, lane = tid & 31;
    const long rowBase = (long)blockIdx.x * 16;

    // LayerNorm: wave w handles rows 2w, 2w+1
    for (int rr = 0; rr < 2; ++rr) {
        const int r = wave * 2 + rr;
        const float* xr = x + (rowBase + r) * CC;
        float vals[8]; float s = 0.f, s2 = 0.f;
        #pragma unroll
        for (int e = 0; e < 8; ++e) {
            float t0 = xr[lane + 32 * e];
            vals[e] = t0; s += t0; s2 += t0 * t0;
        }
        #pragma unroll
        for (int m = 16; m >= 1; m >>= 1) { s += __shfl_xor(s, m, 32); s2 += __shfl_xor(s2, m, 32); }
        const float mu = s * (1.f / CC);
        const float rstd = rsqrtf(s2 * (1.f / CC) - mu * mu + EPSV);
        #pragma unroll
        for (int e = 0; e < 8; ++e) {
            int c0 = lane + 32 * e;
            sA[r * LDA + c0] = (vals[e] - mu) * rstd * nw[c0] + nb[c0];
        }
    }
    __syncthreads();

    const int mrow  = lane & 15;
    const int koff  = (lane >> 4) << 1;   // 0 or 2
    const int mbase = (lane >> 4) << 3;   // 0 or 8

    // 48 (matrix, ntile) pairs over 8 waves -> 6 per wave
    for (int t = 0; t < 6; ++t) {
        const int id = wave * 6 + t;
        const int mat = id >> 4, n0 = (id & 15) * 16;
        const float* W  = (mat == 0) ? wq : ((mat == 1) ? wk : wv);
        const float* Bv_ = (mat == 0) ? bq : ((mat == 1) ? bk : bv);
        float* O = (mat == 0) ? q : ((mat == 1) ? k : v);
        v8f c = {};
        #pragma unroll 4
        for (int kk = 0; kk < CC; kk += 4) {
            v2f a = *(const v2f*)&sA[mrow * LDA + kk + koff];
            v2f b;
            b.x = W[(kk + koff + 0) * CC + n0 + mrow];
            b.y = W[(kk + koff + 1) * CC + n0 + mrow];
            c = wmma_f32(a, b, c);
        }
        const int ncol = n0 + mrow;
        const float bias = Bv_[ncol];
        #pragma unroll
        for (int i = 0; i < 8; ++i)
            O[(rowBase + mbase + i) * CC + ncol] = c[i] + bias;
    }
}

// ---------------- Kernel 2: masked graph attention (per b,t) ----------------
// grid = B*T blocks, 192 threads: thread = (head h, row i). attended overwrites q in place.
__global__ void attn_kernel(float* __restrict__ q, const float* __restrict__ k,
                            const float* __restrict__ v, const int* __restrict__ adj) {
    extern __shared__ float sm[];
    float* kbuf = sm;             // 24*256
    float* vbuf = sm + 24 * CC;   // 24*256
    const int tid = threadIdx.x;  // 0..191
    const long bt = blockIdx.x;
    const float* kg = k + bt * 24 * CC;
    const float* vg = v + bt * 24 * CC;
    #pragma unroll
    for (int e = 0; e < 32; ++e) {
        int idx = tid + 192 * e;
        kbuf[idx] = kg[idx];
        vbuf[idx] = vg[idx];
    }
    __syncthreads();

    const int h = tid / 24, i = tid % 24;
    const int hd = h * 32;
    float* qrow = q + (bt * 24 + i) * CC + hd;
    float qreg[32];
    #pragma unroll
    for (int d = 0; d < 32; ++d) qreg[d] = qrow[d];

    const float scale = 0.1767766952966369f; // 1/sqrt(32)
    float s[24];
    for (int j = 0; j < 24; ++j) {
        float acc = 0.f;
        #pragma unroll
        for (int d = 0; d < 32; ++d) acc += qreg[d] * kbuf[j * CC + hd + d];
        const bool keep = (i == j) || (adj[i * 24 + j] > 0);
        s[j] = keep ? acc * scale : -10000.0f;
    }
    float mmax = -1e30f;
    for (int j = 0; j < 24; ++j) mmax = fmaxf(mmax, s[j]);
    float sum = 0.f;
    for (int j = 0; j < 24; ++j) { s[j] = expf(s[j] - mmax); sum += s[j]; }
    const float inv = 1.0f / sum;
    for (int j = 0; j < 24; ++j) s[j] *= inv;

    #pragma unroll 8
    for (int d = 0; d < 32; ++d) {
        float acc = 0.f;
        for (int j = 0; j < 24; ++j) acc += s[j] * vbuf[j * CC + hd + d];
        qrow[d] = acc;  // attended, in place over q
    }
}

// ---------------- Kernel 3: output projection + residual ----------------
__global__ void oproj_kernel(const float* __restrict__ att, const float* __restrict__ x,
                             const float* __restrict__ wo, const float* __restrict__ bo,
                             float* __restrict__ x2) {
    extern __shared__ float sA[];                 // [16][LDA]
    const int tid = threadIdx.x;
    const int wave = tid >> 5, lane = tid & 31;
    const long rowBase = (long)blockIdx.x * 16;

    #pragma unroll
    for (int u = 0; u < 4; ++u) {
        int sidx = tid * 4 + u;                   // 1024 float4 slots
        int r = sidx >> 6, c4 = (sidx & 63) << 2;
        float4 t4 = *(const float4*)&att[(rowBase + r) * CC + c4];
        *(float4*)&sA[r * LDA + c4] = t4;
    }
    __syncthreads();

    const int mrow  = lane & 15;
    const int koff  = (lane >> 4) << 1;
    const int mbase = (lane >> 4) << 3;

    for (int t = 0; t < 2; ++t) {
        const int n0 = (wave * 2 + t) * 16;
        v8f c = {};
        #pragma unroll 4
        for (int kk = 0; kk < CC; kk += 4) {
            v2f a = *(const v2f*)&sA[mrow * LDA + kk + koff];
            v2f b;
            b.x = wo[(kk + koff + 0) * CC + n0 + mrow];
            b.y = wo[(kk + koff + 1) * CC + n0 + mrow];
            c = wmma_f32(a, b, c);
        }
        const int ncol = n0 + mrow;
        const float bias = bo[ncol];
        #pragma unroll
        for (int i = 0; i < 8; ++i) {
            const long row = rowBase + mbase + i;
            x2[row * CC + ncol] = c[i] + bias + x[row * CC + ncol];
        }
    }
}

// ---------------- Kernel 4: fused FFN (LN -> GEMM -> GELU -> GEMM -> residual) ----
__global__ void ffn_kernel(const float* __restrict__ x2,
                           const float* __restrict__ fw, const float* __restrict__ fb,
                           const float* __restrict__ w1, const float* __restrict__ b1,
                           const float* __restrict__ w2, const float* __restrict__ b2,
                           float* __restrict__ out) {
    extern __shared__ float sm2[];
    float* sA = sm2;                  // [16][LDA]
    float* sG = sm2 + 16 * LDA;       // [16][LDG]
    const int tid = threadIdx.x;
    const int wave = tid >> 5, lane = tid & 31;
    const long rowBase = (long)blockIdx.x * 16;

    // LayerNorm of x2 tile
    for (int rr = 0; rr < 2; ++rr) {
        const int r = wave * 2 + rr;
        const float* xr = x2 + (rowBase + r) * CC;
        float vals[8]; float s = 0.f, s2 = 0.f;
        #pragma unroll
        for (int e = 0; e < 8; ++e) {
            float t0 = xr[lane + 32 * e];
            vals[e] = t0; s += t0; s2 += t0 * t0;
        }
        #pragma unroll
        for (int m = 16; m >= 1; m >>= 1) { s += __shfl_xor(s, m, 32); s2 += __shfl_xor(s2, m, 32); }
        const float mu = s * (1.f / CC);
        const float rstd = rsqrtf(s2 * (1.f / CC) - mu * mu + EPSV);
        #pragma unroll
        for (int e = 0; e < 8; ++e) {
            int c0 = lane + 32 * e;
            sA[r * LDA + c0] = (vals[e] - mu) * rstd * fw[c0] + fb[c0];
        }
    }
    __syncthreads();

    const int mrow  = lane & 15;
    const int koff  = (lane >> 4) << 1;
    const int mbase = (lane >> 4) << 3;

    // Phase 2: h @ w1 + b1 -> exact GELU -> sG  (64 N-tiles over 8 waves)
    for (int t = 0; t < 8; ++t) {
        const int n0 = (wave * 8 + t) * 16;
        v8f c = {};
        #pragma unroll 4
        for (int kk = 0; kk < CC; kk += 4) {
            v2f a = *(const v2f*)&sA[mrow * LDA + kk + koff];
            v2f b;
            b.x = w1[(kk + koff + 0) * 1024 + n0 + mrow];
            b.y = w1[(kk + koff + 1) * 1024 + n0 + mrow];
            c = wmma_f32(a, b, c);
        }
        const int ncol = n0 + mrow;
        const float bias = b1[ncol];
        #pragma unroll
        for (int i = 0; i < 8; ++i) {
            float hv = c[i] + bias;
            sG[(mbase + i) * LDG + ncol] = 0.5f * hv * (1.0f + erff(hv * 0.70710678118654752f));
        }
    }
    __syncthreads();

    // Phase 3: g @ w2 + b2 + x2 -> out  (16 N-tiles over 8 waves)
    for (int t = 0; t < 2; ++t) {
        const int n0 = (wave * 2 + t) * 16;
        v8f c = {};
        #pragma unroll 4
        for (int kk = 0; kk < 1024; kk += 4) {
            v2f a = *(const v2f*)&sG[mrow * LDG + kk + koff];
            v2f b;
            b.x = w2[(kk + koff + 0) * CC + n0 + mrow];
            b.y = w2[(kk + koff + 1) * CC + n0 + mrow];
            c = wmma_f32(a, b, c);
        }
        const int ncol = n0 + mrow;
        const float bias = b2[ncol];
        #pragma unroll
        for (int i = 0; i < 8; ++i) {
            const long row = rowBase + mbase + i;
            out[row * CC + ncol] = c[i] + bias + x2[row * CC + ncol];
        }
    }
}

extern "C" void kernel_launch(void* const* d_in, const int* in_sizes, int n_in,
                              void* d_out, int out_size, void* d_ws, size_t ws_size,
                              hipStream_t stream) {
    const float* x  = (const float*)d_in[0];
    const int*   adj = (const int*)d_in[1];
    const float* nw = (const float*)d_in[2];
    const float* nb = (const float*)d_in[3];
    const float* wq = (const float*)d_in[4];  const float* bq = (const float*)d_in[5];
    const float* wk = (const float*)d_in[6];  const float* bk = (const float*)d_in[7];
    const float* wv = (const float*)d_in[8];  const float* bv = (const float*)d_in[9];
    const float* wo = (const float*)d_in[10]; const float* bo = (const float*)d_in[11];
    const float* fw = (const float*)d_in[12]; const float* fb = (const float*)d_in[13];
    const float* w1 = (const float*)d_in[14]; const float* b1 = (const float*)d_in[15];
    const float* w2 = (const float*)d_in[16]; const float* b2 = (const float*)d_in[17];
    float* out = (float*)d_out;

    const long M = 16L * 512 * 24;            // 196608 tokens
    float* q = (float*)d_ws;                  // M*256 floats; later holds 'attended'
    float* k = q + M * CC;                    // M*256 floats; later holds 'x2'
    float* v = k + M * CC;                    // M*256 floats
    const int mtiles = (int)(M / 16);         // 12288

    ln_qkv_kernel<<<mtiles, 256, 16 * LDA * 4, stream>>>(x, nw, nb, wq, bq, wk, bk, wv, bv, q, k, v);
    attn_kernel<<<16 * 512, 192, 2 * 24 * CC * 4, stream>>>(q, k, v, adj);
    oproj_kernel<<<mtiles, 256, 16 * LDA * 4, stream>>>(q, x, wo, bo, k);
    ffn_kernel<<<mtiles, 256, (16 * LDA + 16 * LDG) * 4, stream>>>(k, fw, fb, w1, b1, w2, b2, out);
}